// MGARCH_DCC_2731599200709
// MI455X (gfx1250) — compile-verified
//
#include <hip/hip_runtime.h>
#include <hip/hip_bf16.h>

// ---------------------------------------------------------------------------
// DCC-GARCH conditional moments on MI455X (gfx1250).
//
// Strategy (output-bandwidth-bound problem: 134MB of Sigmas ~ 6us at 23.3TB/s):
//  K1: mus[t][m] = MU[m]                       (fully parallel)
//  K2: A0 = L0^T L0                            (fully parallel)
//  K3: sig/ustd scan over T (per-series, 64 threads, 1-fma critical path)
//  K4: Q boundary pass: 1 WG, 16 waves, each wave keeps its 16x16 Q tile in
//      8 VGPRs; rank-1 DCC update done with v_wmma_f32_16x16x4_f32; stores
//      exact Q snapshots every CHUNK steps. No barriers (tiles independent).
//  K5: 128 WGs replay their chunk from the exact boundary Q (same WMMA
//      update), extract diag via the f32 C/D lane layout, renormalize, and
//      stream Sigma_t to HBM from the whole chip.
//
// Hot-loop note: v-vector operand loads are done by ALL lanes with a masked
// index (lane & 15) and then selected to zero for lanes 16..31 via cndmask,
// so the serial WMMA chain has no EXEC-mask save/restore or divergent
// branches per step.
// ---------------------------------------------------------------------------

typedef float v2f __attribute__((ext_vector_type(2)));
typedef float v8f __attribute__((ext_vector_type(8)));

#define T_STEPS 8192
#define MDIM    64
#define CHUNK   64
#define NCHUNK  (T_STEPS / CHUNK)   // 128

// workspace layout (float offsets), total ~6.2 MB
#define WS_USTD 0
#define WS_SIG  (T_STEPS * MDIM)                  // 524288
#define WS_A0   (2 * T_STEPS * MDIM)              // 1048576
#define WS_QS   (WS_A0 + MDIM * MDIM)             // + NCHUNK*4096 floats

// -------------------------------- K1: mus ----------------------------------
__global__ void dcc_mus(const float* __restrict__ MU, float* __restrict__ out) {
    int i = blockIdx.x * blockDim.x + threadIdx.x;
    if (i < T_STEPS * MDIM) out[i] = MU[i & (MDIM - 1)];
}

// -------------------------------- K2: A0 = L0^T L0 -------------------------
__global__ void dcc_a0(const float* __restrict__ L0, float* __restrict__ A0) {
    int tid = blockIdx.x * blockDim.x + threadIdx.x;   // 4096 threads total
    int row = tid >> 6, col = tid & 63;
    float acc = 0.0f;
#pragma unroll 4
    for (int k = 0; k < MDIM; ++k)
        acc = fmaf(L0[k * MDIM + row], L0[k * MDIM + col], acc);
    A0[tid] = acc;
}

// -------------------------------- K3: sig / ustd scan ----------------------
// Per-series GARCH(1,1) variance recurrence. u_t = y_t - MU is input-only,
// so the only serial dependency is one fma per step.
__global__ void dcc_sigscan(const float* __restrict__ y,
                            const float* __restrict__ MU,
                            const float* __restrict__ sigma0,
                            const float* __restrict__ alpha0,
                            const float* __restrict__ alpha,
                            const float* __restrict__ beta,
                            float* __restrict__ ustd_o,
                            float* __restrict__ sig_o) {
    int m = threadIdx.x;                  // 64 threads, one per series
    float u   = 0.0f;
    float sig = sigma0[m];
    float a0v = alpha0[m] * alpha0[m];
    float al  = alpha[m], be = beta[m], mu = MU[m];
    for (int t = 0; t < T_STEPS; ++t) {
        float s2 = a0v + al * sig * sig + be * u * u;
        float st = sqrtf(s2);
        ustd_o[t * MDIM + m] = u / sig;   // uses carry (u_{t-1}, sig_{t-1})
        sig_o [t * MDIM + m] = st;
        u   = y[t * MDIM + m] - mu;       // new carry
        sig = st;
    }
}

// ------------- WMMA rank-1 DCC update (shared by K4 / K5) ------------------
// One 16x16 tile per wave.  A-operand K=0 column = Bsc*v[row block],
// B-operand K=0 row = v[col block], all other K slots zero, so
//   D = C + Bsc * outer(v_row, v_col)
// with C = (1-A-B)*A0 + A*Q precomputed by VALU fmas.  K=0 of both operands
// lives in lanes 0..15 of the first operand VGPR per the ISA 16x4/4x16 f32
// layouts; lanes 16..31 and the second VGPR are zeroed (branch-free).
__device__ __forceinline__ v8f dcc_q_step(v8f q, const v8f& ka0,
                                          const float* __restrict__ vrow,
                                          float Asc, float Bsc,
                                          int bi, int bj, int lane) {
    const int li = lane & 15;                  // duplicated load, no divergence
    float va = vrow[16 * bi + li];
    float vb = vrow[16 * bj + li];
    bool lo = (lane < 16);
    v2f a; a.x = lo ? (Bsc * va) : 0.0f; a.y = 0.0f;
    v2f b; b.x = lo ?  vb        : 0.0f; b.y = 0.0f;
    v8f c;
#pragma unroll
    for (int r = 0; r < 8; ++r) c[r] = fmaf(Asc, q[r], ka0[r]);
    return __builtin_amdgcn_wmma_f32_16x16x4_f32(
        /*neg_a=*/false, a, /*neg_b=*/false, b,
        /*c_mod=*/(short)0, c, /*reuse_a=*/false, /*reuse_b=*/false);
}

// -------------------------------- K4: Q boundary pass ----------------------
__global__ __launch_bounds__(512)
void dcc_qbound(const float* __restrict__ ustd,
                const float* __restrict__ A0,
                const float* __restrict__ Ap, const float* __restrict__ Bp,
                float* __restrict__ Qs) {
    const int tid = threadIdx.x;
    const int lane = tid & 31, wv = tid >> 5;
    const int bi = wv >> 2, bj = wv & 3;
    const int hi = lane >> 4, nl = lane & 15;
    const float Asc = Ap[0], Bsc = Bp[0];
    const float Ksc = 1.0f - Asc - Bsc;

    v8f q, ka0;
#pragma unroll
    for (int r = 0; r < 8; ++r) {
        int Mg = 16 * bi + r + 8 * hi;     // f32 C/D layout: VGPR r -> M=r(+8)
        int Ng = 16 * bj + nl;
        float a0v = A0[Mg * MDIM + Ng];
        q[r]   = a0v;                      // initial carry Q = A0
        ka0[r] = Ksc * a0v;
    }

    for (int c = 0; c < NCHUNK; ++c) {
        // snapshot the carry Q entering this chunk (branch-free inner loop)
#pragma unroll
        for (int r = 0; r < 8; ++r) {
            int Mg = 16 * bi + r + 8 * hi;
            Qs[c * (MDIM * MDIM) + Mg * MDIM + 16 * bj + nl] = q[r];
        }
        const float* vbase = ustd + (size_t)c * CHUNK * MDIM;
        for (int s = 0; s < CHUNK; ++s) {
            const float* vrow = vbase + s * MDIM;
            __builtin_prefetch(vrow + 8 * MDIM, 0, 1);   // global_prefetch_b8
            q = dcc_q_step(q, ka0, vrow, Asc, Bsc, bi, bj, lane);
        }
    }
}

// -------------------------------- K5: chunk expand -------------------------
__global__ __launch_bounds__(512)
void dcc_expand(const float* __restrict__ ustd,
                const float* __restrict__ sigt,
                const float* __restrict__ A0,
                const float* __restrict__ Qs,
                const float* __restrict__ Ap, const float* __restrict__ Bp,
                float* __restrict__ Sig) {
    __shared__ float sd[MDIM];   // diag(Q_t)
    __shared__ float sw[MDIM];   // w[m] = sig_t[m] / sqrt(diag Q_t[m])
    const int tid = threadIdx.x;
    const int lane = tid & 31, wv = tid >> 5;
    const int bi = wv >> 2, bj = wv & 3;
    const int hi = lane >> 4, nl = lane & 15;
    const int c = blockIdx.x;
    const float Asc = Ap[0], Bsc = Bp[0];
    const float Ksc = 1.0f - Asc - Bsc;

    v8f q, ka0;
#pragma unroll
    for (int r = 0; r < 8; ++r) {
        int Mg = 16 * bi + r + 8 * hi;
        int Ng = 16 * bj + nl;
        q[r]   = Qs[c * (MDIM * MDIM) + Mg * MDIM + Ng];  // exact boundary Q
        ka0[r] = Ksc * A0[Mg * MDIM + Ng];
    }

    const int t0 = c * CHUNK;
    for (int s = 0; s < CHUNK; ++s) {
        const int t = t0 + s;
        q = dcc_q_step(q, ka0, ustd + (size_t)t * MDIM, Asc, Bsc, bi, bj, lane);

        // diagonal tiles publish diag(Q_t): element (M,N) with M==N sits at
        // VGPR r, lane r (M=r) and lane r+24 (M=r+8) per the f32 C/D layout.
        if (bi == bj) {
#pragma unroll
            for (int r = 0; r < 8; ++r) {
                int Ml = r + 8 * hi;
                if (nl == Ml) sd[16 * bi + Ml] = q[r];
            }
        }
        __syncthreads();
        if (tid < MDIM)
            sw[tid] = sigt[(size_t)t * MDIM + tid] / sqrtf(sd[tid]);
        __syncthreads();

        float wn = sw[16 * bj + nl];
        float* orow = Sig + (size_t)t * (MDIM * MDIM);
#pragma unroll
        for (int r = 0; r < 8; ++r) {
            int Mg = 16 * bi + r + 8 * hi;
            float wm = sw[Mg];
            orow[Mg * MDIM + 16 * bj + nl] = q[r] * wm * wn;
        }
    }
}

// ---------------------------------------------------------------------------
extern "C" void kernel_launch(void* const* d_in, const int* in_sizes, int n_in,
                              void* d_out, int out_size, void* d_ws, size_t ws_size,
                              hipStream_t stream) {
    const float* y      = (const float*)d_in[0];
    const float* MU     = (const float*)d_in[1];
    const float* sigma0 = (const float*)d_in[2];
    const float* alpha0 = (const float*)d_in[3];
    const float* alpha  = (const float*)d_in[4];
    const float* beta   = (const float*)d_in[5];
    const float* L0     = (const float*)d_in[6];
    const float* A      = (const float*)d_in[7];
    const float* B      = (const float*)d_in[8];

    float* out  = (float*)d_out;
    float* wsf  = (float*)d_ws;
    float* wU   = wsf + WS_USTD;
    float* wS   = wsf + WS_SIG;
    float* wA0  = wsf + WS_A0;
    float* wQs  = wsf + WS_QS;

    dcc_mus<<<(T_STEPS * MDIM + 255) / 256, 256, 0, stream>>>(MU, out);
    dcc_a0<<<16, 256, 0, stream>>>(L0, wA0);
    dcc_sigscan<<<1, 64, 0, stream>>>(y, MU, sigma0, alpha0, alpha, beta, wU, wS);
    dcc_qbound<<<1, 512, 0, stream>>>(wU, wA0, A, B, wQs);
    dcc_expand<<<NCHUNK, 512, 0, stream>>>(wU, wS, wA0, wQs, A, B,
                                           out + T_STEPS * MDIM);
}